// MultiHeadSelfAttention_23768349016249
// MI455X (gfx1250) — compile-verified
//
#include <hip/hip_runtime.h>
#include <hip/hip_bf16.h>
#include <math.h>

// ---------------------------------------------------------------------------
// MultiHeadSelfAttention forward for MI455X (gfx1250), fp32 end-to-end using
// V_WMMA_F32_16X16X4_F32 for all matrix math.
//   B=2, T=2048, D_MODEL=1024, N_HEADS=16, HEAD_DIM=64
// ---------------------------------------------------------------------------

#define BB      2
#define TT      2048
#define DM      1024
#define NH      16
#define DH      64
#define ROWS    (BB * TT)        // 4096
#define QKV_N   (3 * DM)         // 3072

typedef __attribute__((ext_vector_type(2))) float v2f;
typedef __attribute__((ext_vector_type(8))) float v8f;

// ---------------------------------------------------------------------------
// Generic GEMM + bias:  C[M x N] = A[M x K] * Bm[K x N] + bias[N]
// One wave computes a 32(M) x 64(N) strip: 8 accumulators; each strided
// B-fragment feeds two WMMAs (A-frag reuse across M sub-tiles).
// All K-loop addresses are carried pointers (uniform strides) so the loop is
// just pointer bumps + loads + WMMAs.
// f32 WMMA fragment layouts (wave32):
//   A 16x4 : lane m=l&15, half=l>>4 ; a[v] = A[m][k0 + v + 2*half]  (float2)
//   B 4x16 : n=l&15                ; b[v] = B[k0 + v + 2*half][n]
//   C 16x16: vgpr v, row = v + 8*half, col = l&15
// ---------------------------------------------------------------------------
__global__ void gemm_bias_f32wmma(const float* __restrict__ A,
                                  const float* __restrict__ Bm,
                                  const float* __restrict__ bias,
                                  float* __restrict__ C,
                                  int M, int N, int K)
{
    const int wave = blockIdx.x * (blockDim.x >> 5) + (threadIdx.x >> 5);
    const int lane = threadIdx.x & 31;
    const int half = lane >> 4;
    const int ln   = lane & 15;

    const int ngroups = N >> 6;               // 64-wide column groups
    const int mt = wave / ngroups;            // 32-row group
    const int ng = wave % ngroups;
    if (mt * 32 >= M) return;                 // wave-uniform

    const float* __restrict__ ap0 = A + (size_t)(mt * 32 + ln) * K + 2 * half;
    const float* __restrict__ ap1 = ap0 + (size_t)16 * K;
    const float* __restrict__ bp  = Bm + (size_t)(2 * half) * N + ng * 64 + ln;

    v8f acc00 = {}, acc01 = {}, acc02 = {}, acc03 = {};
    v8f acc10 = {}, acc11 = {}, acc12 = {}, acc13 = {};

    for (int k0 = 0; k0 < K; k0 += 4) {
        const v2f a0 = *(const v2f*)ap0;
        const v2f a1 = *(const v2f*)ap1;

        v2f b0; b0.x = bp[0];  b0.y = bp[N];
        v2f b1; b1.x = bp[16]; b1.y = bp[N + 16];
        v2f b2; b2.x = bp[32]; b2.y = bp[N + 32];
        v2f b3; b3.x = bp[48]; b3.y = bp[N + 48];

        acc00 = __builtin_amdgcn_wmma_f32_16x16x4_f32(false, a0, false, b0, (short)0, acc00, false, false);
        acc10 = __builtin_amdgcn_wmma_f32_16x16x4_f32(false, a1, false, b0, (short)0, acc10, false, false);
        acc01 = __builtin_amdgcn_wmma_f32_16x16x4_f32(false, a0, false, b1, (short)0, acc01, false, false);
        acc11 = __builtin_amdgcn_wmma_f32_16x16x4_f32(false, a1, false, b1, (short)0, acc11, false, false);
        acc02 = __builtin_amdgcn_wmma_f32_16x16x4_f32(false, a0, false, b2, (short)0, acc02, false, false);
        acc12 = __builtin_amdgcn_wmma_f32_16x16x4_f32(false, a1, false, b2, (short)0, acc12, false, false);
        acc03 = __builtin_amdgcn_wmma_f32_16x16x4_f32(false, a0, false, b3, (short)0, acc03, false, false);
        acc13 = __builtin_amdgcn_wmma_f32_16x16x4_f32(false, a1, false, b3, (short)0, acc13, false, false);

        ap0 += 4;
        ap1 += 4;
        bp  += (size_t)4 * N;   // uniform stride: strength-reduced pointer bump
    }

    // Epilogue: C row = mt*32 + sub*16 + v + 8*half, col = ng*64 + t*16 + ln
    v8f accs[2][4] = { { acc00, acc01, acc02, acc03 },
                       { acc10, acc11, acc12, acc13 } };
#pragma unroll
    for (int sub = 0; sub < 2; ++sub) {
        const int rbase = mt * 32 + sub * 16 + 8 * half;
#pragma unroll
        for (int t = 0; t < 4; ++t) {
            const int col = ng * 64 + t * 16 + ln;
            const float bs = bias[col];
#pragma unroll
            for (int v = 0; v < 8; ++v)
                C[(size_t)(rbase + v) * N + col] = accs[sub][t][v] + bs;
        }
    }
}

// ---------------------------------------------------------------------------
// RoPE + head split: qkv (ROWS x 3072) -> qb/kb/vb each (B, H, T, DH)
// ---------------------------------------------------------------------------
__global__ void rope_split(const float* __restrict__ qkv,
                           float* __restrict__ qb,
                           float* __restrict__ kb,
                           float* __restrict__ vb)
{
    const int row = blockIdx.x;          // b*T + t
    const int b = row / TT;
    const int t = row % TT;
    const float* __restrict__ src = qkv + (size_t)row * QKV_N;

    for (int p = threadIdx.x; p < QKV_N / 2; p += blockDim.x) {
        const int c   = p * 2;           // even column
        const int sec = c >> 10;         // 0=q, 1=k, 2=v
        const int cc  = c & (DM - 1);
        const int h   = cc >> 6;
        const int d   = cc & (DH - 1);

        const float e = src[c];
        const float o = src[c + 1];
        const size_t dst = (((size_t)(b * NH + h)) * TT + t) * DH + d;

        if (sec == 2) {
            vb[dst]     = e;
            vb[dst + 1] = o;
        } else {
            const int   i   = d >> 1;
            // inv_freq = 10000^(-2i/64)
            const float inv = expf(-(float)(2 * i) * (9.210340371976184f / 64.0f));
            const float ang = (float)t * inv;
            const float sn = sinf(ang);
            const float cs = cosf(ang);
            const float re = e * cs - o * sn;
            const float ro = e * sn + o * cs;
            if (sec == 0) { qb[dst] = re; qb[dst + 1] = ro; }
            else          { kb[dst] = re; kb[dst + 1] = ro; }
        }
    }
}

// ---------------------------------------------------------------------------
// Flash attention (causal + padding mask), fp32 WMMA.
// Grid: B*H*(T/64) blocks of 128 threads (4 waves). Each wave owns 16 q-rows,
// streams causal key tiles of 16, online softmax, P routed through padded LDS.
// ---------------------------------------------------------------------------
#define PSTRIDE 20   // bank-conflict padding: halves land 32 banks apart

__global__ void flash_attn_f32wmma(const float* __restrict__ qb,
                                   const float* __restrict__ kb,
                                   const float* __restrict__ vb,
                                   const int*   __restrict__ amask,
                                   float* __restrict__ attn /* (B*T) x DM */)
{
    __shared__ float pshare[4][16 * PSTRIDE];

    const int wid  = threadIdx.x >> 5;
    const int lane = threadIdx.x & 31;
    const int half = lane >> 4;
    const int ln   = lane & 15;

    const int blocksPerBH = TT / 64;                 // 32
    const int bh   = blockIdx.x / blocksPerBH;       // 0..B*NH-1
    const int qblk = blockIdx.x % blocksPerBH;
    const int b    = bh >> 4;                        // NH == 16
    const int q0   = qblk * 64 + wid * 16;           // wave's q-row base

    const float* __restrict__ Q  = qb + (size_t)bh * TT * DH;
    const float* __restrict__ Kp = kb + (size_t)bh * TT * DH;
    const float* __restrict__ Vp = vb + (size_t)bh * TT * DH;

    // Preload this wave's 16x64 Q tile as 16 A-fragments (float2 each).
    v2f qf[16];
    {
        const float* __restrict__ qrow = Q + (size_t)(q0 + ln) * DH;
#pragma unroll
        for (int i = 0; i < 16; ++i)
            qf[i] = *(const v2f*)(qrow + 4 * i + 2 * half);
    }

    v8f o0 = {}, o1 = {}, o2 = {}, o3 = {};
    float rmax[8], rsum[8];
#pragma unroll
    for (int v = 0; v < 8; ++v) { rmax[v] = -1e30f; rsum[v] = 0.0f; }

    float* __restrict__ P = &pshare[wid][0];
    const float scale = 0.125f;                      // 1/sqrt(64)
    const int nkt = q0 / 16 + 1;                     // causal tile count

    for (int j = 0; j < nkt; ++j) {
        const int kbase = j * 16;

        // ---- S = Q * K^T (16x16), 16 f32-WMMAs over Dh=64 ----
        v8f s = {};
        {
            const float* __restrict__ krow = Kp + (size_t)(kbase + ln) * DH;
#pragma unroll
            for (int i = 0; i < 16; ++i) {
                const v2f bf = *(const v2f*)(krow + 4 * i + 2 * half);
                s = __builtin_amdgcn_wmma_f32_16x16x4_f32(false, qf[i], false, bf,
                                                          (short)0, s, false, false);
            }
        }

        // ---- scale + causal/pad mask + online softmax (per C-row) ----
        const int keyIdx = kbase + ln;               // this lane's key column
        const int pm     = amask[b * TT + keyIdx];
#pragma unroll
        for (int v = 0; v < 8; ++v) {
            const int qrowi = q0 + v + 8 * half;     // this element's q row
            float val = s[v] * scale;
            val = (keyIdx > qrowi || pm == 0) ? -1e30f : val;

            // row max over the 16 lanes holding this row
            float mx = val;
            mx = fmaxf(mx, __shfl_xor(mx, 1, 32));
            mx = fmaxf(mx, __shfl_xor(mx, 2, 32));
            mx = fmaxf(mx, __shfl_xor(mx, 4, 32));
            mx = fmaxf(mx, __shfl_xor(mx, 8, 32));

            const float nmax = fmaxf(rmax[v], mx);
            const float p    = __expf(val - nmax);

            float sr = p;
            sr += __shfl_xor(sr, 1, 32);
            sr += __shfl_xor(sr, 2, 32);
            sr += __shfl_xor(sr, 4, 32);
            sr += __shfl_xor(sr, 8, 32);

            const float alpha = __expf(rmax[v] - nmax);
            rsum[v] = rsum[v] * alpha + sr;
            rmax[v] = nmax;
            o0[v] *= alpha; o1[v] *= alpha; o2[v] *= alpha; o3[v] *= alpha;

            P[(v + 8 * half) * PSTRIDE + ln] = p;    // P[m][n], padded
        }

        // Wave-local LDS ordering: DS ops are in-order per wave.
        asm volatile("s_wait_dscnt 0x0" ::: "memory");

        // ---- O += P * V : P as A-frags from LDS, V as B-frags ----
#pragma unroll
        for (int i = 0; i < 4; ++i) {
            const v2f af = *(const v2f*)(P + ln * PSTRIDE + 4 * i + 2 * half);
            const float* __restrict__ vrow =
                Vp + (size_t)(kbase + 4 * i + 2 * half) * DH + ln;
            v2f b0; b0.x = vrow[0];  b0.y = vrow[DH];
            v2f b1; b1.x = vrow[16]; b1.y = vrow[DH + 16];
            v2f b2; b2.x = vrow[32]; b2.y = vrow[DH + 32];
            v2f b3; b3.x = vrow[48]; b3.y = vrow[DH + 48];
            o0 = __builtin_amdgcn_wmma_f32_16x16x4_f32(false, af, false, b0, (short)0, o0, false, false);
            o1 = __builtin_amdgcn_wmma_f32_16x16x4_f32(false, af, false, b1, (short)0, o1, false, false);
            o2 = __builtin_amdgcn_wmma_f32_16x16x4_f32(false, af, false, b2, (short)0, o2, false, false);
            o3 = __builtin_amdgcn_wmma_f32_16x16x4_f32(false, af, false, b3, (short)0, o3, false, false);
        }
    }

    // ---- normalize and scatter to (B, T, H*DH) layout for the out-proj ----
    const int h = bh & (NH - 1);
    v8f os[4] = { o0, o1, o2, o3 };
#pragma unroll
    for (int v = 0; v < 8; ++v) {
        const float inv = 1.0f / rsum[v];
        const size_t row = (size_t)b * TT + (q0 + v + 8 * half);
#pragma unroll
        for (int t = 0; t < 4; ++t)
            attn[row * DM + h * DH + t * 16 + ln] = os[t][v] * inv;
    }
}

// ---------------------------------------------------------------------------
// Host-side launcher.
// Inputs: 0=x, 1=attention_mask(int), 2=W_qkv, 3=b_qkv, 4=W_out, 5=b_out
// Workspace: qkv (ROWS*3072) | qb | kb | vb  (each B*H*T*DH); attn aliases qkv.
// ---------------------------------------------------------------------------
extern "C" void kernel_launch(void* const* d_in, const int* in_sizes, int n_in,
                              void* d_out, int out_size, void* d_ws, size_t ws_size,
                              hipStream_t stream)
{
    (void)in_sizes; (void)n_in; (void)out_size; (void)ws_size;

    const float* x      = (const float*)d_in[0];
    const int*   mask   = (const int*)  d_in[1];
    const float* W_qkv  = (const float*)d_in[2];
    const float* b_qkv  = (const float*)d_in[3];
    const float* W_out  = (const float*)d_in[4];
    const float* b_out  = (const float*)d_in[5];
    float*       out    = (float*)d_out;

    float* ws   = (float*)d_ws;
    float* qkv  = ws;                                   // ROWS * 3072
    float* qb   = qkv + (size_t)ROWS * QKV_N;           // B*H*T*DH
    float* kb   = qb  + (size_t)BB * NH * TT * DH;
    float* vb   = kb  + (size_t)BB * NH * TT * DH;
    float* attn = qkv;                                  // alias: qkv dead by then

    // 1) QKV projection: (4096x1024)*(1024x3072)+bias
    {
        const int waves = (ROWS / 32) * (QKV_N / 64);   // 6144
        gemm_bias_f32wmma<<<waves / 4, 128, 0, stream>>>(
            x, W_qkv, b_qkv, qkv, ROWS, QKV_N, DM);
    }

    // 2) RoPE + split into (B,H,T,DH)
    rope_split<<<ROWS, 256, 0, stream>>>(qkv, qb, kb, vb);

    // 3) Causal flash attention
    {
        const int blocks = BB * NH * (TT / 64);         // 1024
        flash_attn_f32wmma<<<blocks, 128, 0, stream>>>(qb, kb, vb, mask, attn);
    }

    // 4) Output projection: (4096x1024)*(1024x1024)+bias
    {
        const int waves = (ROWS / 32) * (DM / 64);      // 2048
        gemm_bias_f32wmma<<<waves / 4, 128, 0, stream>>>(
            attn, W_out, b_out, out, ROWS, DM, DM);
    }
}